// MultiHeadLatentAttention_80461917323891
// MI455X (gfx1250) — compile-verified
//
#include <hip/hip_runtime.h>
#include <hip/hip_bf16.h>
#include <math.h>

// ---------------------------------------------------------------------------
// MLA forward for MI455X (gfx1250, wave32). All matrix math on the CDNA5
// f32 WMMA path: V_WMMA_F32_16X16X4_F32 (D = A(16x4) * B(4x16) + C(16x16)).
//
// f32 WMMA operand layout (ISA 7.12.2):
//   A (16x4):  lanes 0-15 -> M=lane,   vgpr0=K0, vgpr1=K1
//              lanes16-31 -> M=lane-16,vgpr0=K2, vgpr1=K3
//   B (4x16):  lanes 0-15 -> N=lane,   vgpr0=K0, vgpr1=K1
//              lanes16-31 -> N=lane-16,vgpr0=K2, vgpr1=K3
//   C/D (16x16): vgpr r, lanes 0-15 -> M=r, N=lane; lanes 16-31 -> M=r+8.
//
// All WMMA operands are read from LDS as contiguous 8B-aligned v2f pairs so
// every fragment is exactly one ds_load_b64 into an even VGPR pair (no
// repacking movs). B-operand tiles are stored pair-interleaved over the WMMA
// K dimension: element (k,n) at lds[(k>>1)*BSTR + 2n + (k&1)].
// ---------------------------------------------------------------------------

typedef float v2f __attribute__((ext_vector_type(2)));
typedef float v8f __attribute__((ext_vector_type(8)));

__device__ __forceinline__ v8f wmma_f32(v2f a, v2f b, v8f c) {
  // (neg_a, A, neg_b, B, c_mod, C, reuse_a, reuse_b)
  return __builtin_amdgcn_wmma_f32_16x16x4_f32(false, a, false, b, (short)0, c,
                                               false, false);
}

#define TSTR 68    // row-major tile stride (A/K/Q tiles): float4-aligned,
                   // fragment banks (4*lane + c) conflict-free
#define BSTR 132   // pair-interleaved stride (B/V tiles): float4-aligned

// ---------------------------------------------------------------------------
// f32 GEMM: Cout[M,N] = A[M,K] @ B[K,N] (+ bias[N]).
// Block = 128 threads (4 waves). Block tile = 64(M) x 64(N); wave w owns the
// 16-row strip m = m0+16w, all 64 columns (4 accumulators). A row-major and
// B pair-interleaved in LDS, staged with coalesced vector loads; the fully
// unrolled inner loop is pure ds_load_b64 + WMMA at immediate offsets.
// Requires M%64==0, N%64==0, K%64==0 (true for all calls here).
// ---------------------------------------------------------------------------
__global__ __launch_bounds__(128) void gemm16_f32(
    const float* __restrict__ A, const float* __restrict__ B,
    float* __restrict__ Cout, const float* __restrict__ bias,
    int M, int N, int K) {
  __shared__ float lds_a[64 * TSTR];   // 64 rows(M) x 64 cols(K), row-major
  __shared__ float lds_b[32 * BSTR];   // 32 k-pairs x 64 cols(N), interleaved

  const int m0   = blockIdx.x * 64;
  const int n0   = blockIdx.y * 64;
  const int tid  = threadIdx.x;
  const int wave = tid >> 5;
  const int lane = tid & 31;
  const int half = lane >> 4;          // 0: K0/K1 group, 1: K2/K3 group
  const int l16  = lane & 15;
  const int wm   = wave * 16;          // wave's M offset inside block tile

  v8f acc[4] = {};

  for (int k0 = 0; k0 < K; k0 += 64) {
    for (int i = 0; i < 8; ++i) {
      int idx = i * 128 + tid;         // 0..1023
      // A: row-major, one b128 per unit, coalesced.
      int ra = idx >> 4, ca = idx & 15;
      *(float4*)&lds_a[ra * TSTR + ca * 4] =
          *(const float4*)(A + (size_t)(m0 + ra) * K + k0 + ca * 4);
      // B: pair-interleave rows k=2p,2p+1 -> {g0.x,g1.x,g0.y,g1.y}.
      int p = idx >> 5, np = idx & 31; // k-pair 0..31, n-pair 0..31
      const float* bp = B + (size_t)(k0 + 2 * p) * N + n0 + 2 * np;
      float2 g0 = *(const float2*)(bp);
      float2 g1 = *(const float2*)(bp + N);
      float4 w; w.x = g0.x; w.y = g1.x; w.z = g0.y; w.w = g1.y;
      *(float4*)&lds_b[p * BSTR + 4 * np] = w;
    }
    __syncthreads();

#pragma unroll
    for (int kk = 0; kk < 64; kk += 4) {
      v2f af = *(const v2f*)&lds_a[(wm + l16) * TSTR + kk + half * 2];
#pragma unroll
      for (int c = 0; c < 4; ++c) {
        v2f bf =
            *(const v2f*)&lds_b[(kk / 2 + half) * BSTR + (c * 16 + l16) * 2];
        acc[c] = wmma_f32(af, bf, acc[c]);
      }
    }
    __syncthreads();
  }

  for (int c = 0; c < 4; ++c) {
    const float bv = bias ? bias[n0 + c * 16 + l16] : 0.0f;
    for (int r = 0; r < 8; ++r) {
      int row = m0 + wm + r + 8 * half;
      Cout[(size_t)row * N + n0 + c * 16 + l16] = acc[c][r] + bv;
    }
  }
}

// ---------------------------------------------------------------------------
// Fused flash attention with additive interaction bias, causal + padding mask.
// grid = (T/16, H, B), one wave per 16-query tile of one (b,h).
// Q/K/V live in (B,T,H*D) row-major buffers. Streams interaction_matrix once;
// prefetches the next key tile's interaction lines (each 64B line serves all
// 16 heads of a (k,q) pair out of the 192MB L2).
// ---------------------------------------------------------------------------
__global__ __launch_bounds__(32) void mla_attention(
    const float* __restrict__ Qm, const float* __restrict__ Km,
    const float* __restrict__ Vm, const float* __restrict__ inter,
    const int* __restrict__ pad, float* __restrict__ ctx) {
  constexpr int T_ = 1024, H_ = 16, C_ = 1024, D_ = 64;
  __shared__ float lds_k[16 * TSTR];   // K tile row-major (Q tile in setup)
  __shared__ float lds_v[8 * BSTR];    // V tile, pair-interleaved over keys
  __shared__ float lds_p[16 * 16];     // P tile: C-layout -> A-layout

  const int qt = blockIdx.x, h = blockIdx.y, b = blockIdx.z;
  const int q0   = qt * 16;
  const int lane = threadIdx.x;
  const int half = lane >> 4;
  const int l16  = lane & 15;
  const float scale = 0.125f;          // D^-0.5, D = 64
  const float NEG_INF = -__builtin_inff();

  // ---- Stage Q tile via LDS (coalesced), extract 16 K=4 A-slices. ----
  v2f qf[16];
  {
    for (int i = 0; i < 8; ++i) {
      int idx = i * 32 + lane;         // 0..255
      int r = idx >> 4, c4 = idx & 15;
      *(float4*)&lds_k[r * TSTR + c4 * 4] = *(const float4*)(
          Qm + ((size_t)b * T_ + q0 + r) * C_ + (size_t)h * D_ + c4 * 4);
    }
    __syncthreads();
#pragma unroll
    for (int kk = 0; kk < 16; ++kk)
      qf[kk] = *(const v2f*)&lds_k[l16 * TSTR + kk * 4 + half * 2];
    __syncthreads();
  }

  float m_row[8], l_row[8];
  for (int r = 0; r < 8; ++r) { m_row[r] = NEG_INF; l_row[r] = 0.0f; }
  v8f O[4] = {};                       // 16 queries x 64 dims accumulator

  for (int kt = 0; kt <= qt; ++kt) {
    const int k0 = kt * 16;

    // Stage K (row-major) and V (pair-interleaved over keys), coalesced.
    for (int i = 0; i < 8; ++i) {
      int idx = i * 32 + lane;
      int r = idx >> 4, c4 = idx & 15;
      *(float4*)&lds_k[r * TSTR + c4 * 4] = *(const float4*)(
          Km + ((size_t)b * T_ + k0 + r) * C_ + (size_t)h * D_ + c4 * 4);
      int p = idx >> 5, np = idx & 31; // key-pair 0..7, d-pair 0..31
      const float* vp =
          Vm + ((size_t)b * T_ + k0 + 2 * p) * C_ + (size_t)h * D_ + 2 * np;
      float2 g0 = *(const float2*)(vp);
      float2 g1 = *(const float2*)(vp + C_);
      float4 w; w.x = g0.x; w.y = g1.x; w.z = g0.y; w.w = g1.y;
      *(float4*)&lds_v[p * BSTR + 4 * np] = w;
    }

    // Prefetch next tile's interaction lines while this tile computes.
    if (kt < qt) {
      const float* nx =
          inter + (((size_t)b * T_ + (k0 + 16 + l16)) * T_ + q0) * H_ + h;
#pragma unroll
      for (int r = 0; r < 8; r += 2)
        __builtin_prefetch(nx + (size_t)(r + 8 * half) * H_, 0, 0);
    }
    __syncthreads();

    // S = Q * K^T  (16 WMMAs over d). B-fragment: n = key = k0 + l16,
    // contiguous d-pair in lds_k row -> single ds_load_b64.
    v8f s = {};
#pragma unroll
    for (int kk = 0; kk < 16; ++kk) {
      v2f bf = *(const v2f*)&lds_k[l16 * TSTR + kk * 4 + half * 2];
      s = wmma_f32(qf[kk], bf, s);
    }

    // scale -> padding replace -> causal replace -> + interaction[b,k,q,h]
    float sv[8];
    for (int r = 0; r < 8; ++r) {
      int M = r + 8 * half;
      int qi = q0 + M, ki = k0 + l16;
      float v = s[r] * scale;
      if (pad[(size_t)b * T_ + qi] == 0) v = -1.0e9f;
      if (ki > qi) v = NEG_INF;
      v += inter[(((size_t)b * T_ + ki) * T_ + qi) * H_ + h];
      sv[r] = v;
    }

    // Online softmax (rows reduce across the 16-lane N groups).
    float alpha[8];
    for (int r = 0; r < 8; ++r) {
      float mx = sv[r];
      for (int off = 1; off < 16; off <<= 1)
        mx = fmaxf(mx, __shfl_xor(mx, off, 32));
      float mnew = fmaxf(m_row[r], mx);  // finite: diagonal always valid
      float p = __expf(sv[r] - mnew);
      float su = p;
      for (int off = 1; off < 16; off <<= 1)
        su += __shfl_xor(su, off, 32);
      float a = __expf(m_row[r] - mnew); // first tile: exp(-inf)=0
      l_row[r] = l_row[r] * a + su;
      m_row[r] = mnew;
      alpha[r] = a;
      sv[r] = p;
    }

    // Transpose P (C-layout) -> A-layout through LDS.
    for (int r = 0; r < 8; ++r)
      lds_p[(r + 8 * half) * 16 + l16] = sv[r];
    __syncthreads();   // single-wave WG: S_NOP; DS stays in-order in-wave

    // Rescale running accumulator, then O += P @ V (16 WMMAs).
    for (int c = 0; c < 4; ++c)
      for (int r = 0; r < 8; ++r) O[c][r] *= alpha[r];

#pragma unroll
    for (int kk = 0; kk < 4; ++kk) {     // key slices of 4
      v2f pf = *(const v2f*)&lds_p[l16 * 16 + kk * 4 + half * 2];
#pragma unroll
      for (int c = 0; c < 4; ++c) {      // d chunks of 16
        v2f vf =
            *(const v2f*)&lds_v[(kk * 2 + half) * BSTR + (c * 16 + l16) * 2];
        O[c] = wmma_f32(pf, vf, O[c]);
      }
    }
    __syncthreads();
  }

  // Normalize and scatter into ctx (B,T,H*D) layout.
  for (int c = 0; c < 4; ++c)
    for (int r = 0; r < 8; ++r) {
      int qi = q0 + r + 8 * half;
      ctx[((size_t)b * T_ + qi) * C_ + (size_t)h * D_ + c * 16 + l16] =
          O[c][r] / l_row[r];
    }
}

// ---------------------------------------------------------------------------
extern "C" void kernel_launch(void* const* d_in, const int* in_sizes, int n_in,
                              void* d_out, int out_size, void* d_ws,
                              size_t ws_size, hipStream_t stream) {
  (void)in_sizes; (void)n_in; (void)out_size; (void)ws_size;
  const float* x      = (const float*)d_in[0];
  const int*   pad    = (const int*)d_in[1];
  const float* inter  = (const float*)d_in[2];
  const float* W_ckv  = (const float*)d_in[3];
  const float* W_cq   = (const float*)d_in[4];
  const float* W_kc   = (const float*)d_in[5];
  const float* W_qc   = (const float*)d_in[6];
  const float* W_vc   = (const float*)d_in[7];
  const float* W_proj = (const float*)d_in[8];
  const float* b_proj = (const float*)d_in[9];
  float* out = (float*)d_out;

  const int B = 4, T = 1024, C = 1024, H = 16, L = 256;
  const size_t BT = (size_t)B * T;

  float* ws      = (float*)d_ws;
  float* kv_down = ws;                       // BT x L
  float* q_down  = kv_down + BT * L;         // BT x L
  float* Kmat    = q_down + BT * L;          // BT x C
  float* Qmat    = Kmat + BT * C;            // BT x C
  float* Vmat    = Qmat + BT * C;            // BT x C
  float* ctxb    = Vmat + BT * C;            // BT x C

  dim3 blk(128);

  // Down-projections: (BT x C) @ (C x L)
  gemm16_f32<<<dim3(BT / 64, L / 64), blk, 0, stream>>>(x, W_ckv, kv_down,
                                                        nullptr, BT, L, C);
  gemm16_f32<<<dim3(BT / 64, L / 64), blk, 0, stream>>>(x, W_cq, q_down,
                                                        nullptr, BT, L, C);
  // Up-projections: (BT x L) @ (L x C)
  gemm16_f32<<<dim3(BT / 64, C / 64), blk, 0, stream>>>(kv_down, W_kc, Kmat,
                                                        nullptr, BT, C, L);
  gemm16_f32<<<dim3(BT / 64, C / 64), blk, 0, stream>>>(q_down, W_qc, Qmat,
                                                        nullptr, BT, C, L);
  gemm16_f32<<<dim3(BT / 64, C / 64), blk, 0, stream>>>(kv_down, W_vc, Vmat,
                                                        nullptr, BT, C, L);
  // Fused causal attention with interaction bias.
  mla_attention<<<dim3(T / 16, H, B), dim3(32), 0, stream>>>(
      Qmat, Kmat, Vmat, inter, pad, ctxb);
  // Output projection with bias.
  gemm16_f32<<<dim3(BT / 64, C / 64), blk, 0, stream>>>(ctxb, W_proj, out,
                                                        b_proj, BT, C, C);
}